// GCN_18743237280158
// MI455X (gfx1250) — compile-verified
//
#include <hip/hip_runtime.h>
#include <cmath>

typedef float v2f __attribute__((ext_vector_type(2)));
typedef float v8f __attribute__((ext_vector_type(8)));

#define N_    256
#define CIN   64
#define COUT  64
#define T_    64
#define V_    25
#define REL   8
#define TV    (T_ * V_)   // 1600
#define VP    28          // V padded to multiple of 4 (WMMA K step); 28 mod 64 banks -> conflict-free
#define UP    32          // U padded to multiple of 16 (WMMA N tile)
#define WRS   68          // wr row stride in LDS: 68 mod 64 = 4 -> lanes hit distinct banks
#define BLK   512         // 16 wave32 -> 4 waves/SIMD for latency hiding
#define NWAVE 16

// ---- LDS layout (in floats) ----
#define OFF_WR    0
#define SZ_WR     (COUT * WRS)            // 4352
#define OFF_XBAR  (OFF_WR + SZ_WR)        // 4352
#define SZ_XBAR   (CIN * V_)              // 1600
#define OFF_X1    (OFF_XBAR + SZ_XBAR)    // 5952
#define OFF_X2    (OFF_X1 + REL * V_)     // 6152
#define OFF_D     (OFF_X2 + REL * V_)     // 6352
#define SZ_D      (REL * V_ * V_)         // 5000
#define OFF_INP   (OFF_D + SZ_D)          // 11352
#define SZ_INP    (16 * T_ * VP)          // 28672
#define OFF_AF    (OFF_INP + SZ_INP)      // 40024
#define SZ_AF     (16 * UP * VP)          // 14336
#define SMEM_FLOATS (OFF_AF + SZ_AF)      // 54360 floats = 217440 bytes < 320 KB/WGP

__global__ void gcn_fused_kernel(const float* __restrict__ x,
                                 const float* __restrict__ Ag,
                                 const float* __restrict__ w1, const float* __restrict__ b1,
                                 const float* __restrict__ w2, const float* __restrict__ b2,
                                 const float* __restrict__ w3, const float* __restrict__ b3,
                                 const float* __restrict__ wr, const float* __restrict__ br,
                                 float* __restrict__ out)
{
    extern __shared__ float sm[];
    const int n    = blockIdx.x;
    const int tid  = threadIdx.x;
    const int wave = tid >> 5;
    const int lane = tid & 31;
    const int lo   = lane & 15;   // row/col within 16-wide half
    const int hi   = lane >> 4;   // which K-pair / M-half
    const float* __restrict__ xg = x + (size_t)n * CIN * TV;

    // ---- Phase 0: stage wr into LDS (bank-conflict-free padded rows); zero padded buffers ----
    for (int idx = tid; idx < COUT * CIN; idx += BLK) {
        const int c = idx / CIN, i = idx % CIN;
        sm[OFF_WR + c * WRS + i] = wr[idx];
    }
    for (int idx = tid; idx < SZ_INP + SZ_AF; idx += BLK) sm[OFF_INP + idx] = 0.0f;

    // ---- Phase 1: xbar[i][v] = mean over t of x[n,i,t,v] ----
    for (int idx = tid; idx < CIN * V_; idx += BLK) {
        const int i = idx / V_, v = idx % V_;
        const float* p = xg + i * TV + v;
        float s = 0.0f;
#pragma unroll 8
        for (int t = 0; t < T_; ++t) s += p[t * V_];
        sm[OFF_XBAR + idx] = s * (1.0f / T_);
    }
    __syncthreads();

    // ---- Phase 2a: x1[r][v], x2[r][v] ----
    for (int idx = tid; idx < 2 * REL * V_; idx += BLK) {
        const int which = idx / (REL * V_);
        const int rv = idx % (REL * V_);
        const int r = rv / V_, v = rv % V_;
        const float* __restrict__ w = which ? w2 : w1;
        float s = which ? b2[r] : b1[r];
#pragma unroll 8
        for (int i = 0; i < CIN; ++i) s += w[r * CIN + i] * sm[OFF_XBAR + i * V_ + v];
        sm[(which ? OFF_X2 : OFF_X1) + rv] = s;
    }
    __syncthreads();

    // ---- Phase 2b: d[r][u][v] = tanh(x1[r][u] - x2[r][v]) ----
    for (int idx = tid; idx < SZ_D; idx += BLK) {
        const int r = idx / (V_ * V_);
        const int uv = idx % (V_ * V_);
        const int u = uv / V_, v = uv % V_;
        sm[OFF_D + idx] = tanhf(sm[OFF_X1 + r * V_ + u] - sm[OFF_X2 + r * V_ + v]);
    }

    // ---- Channel tiles of 16 ----
    for (int ct = 0; ct < COUT / 16; ++ct) {
        __syncthreads();  // prev iter's GEMM2 reads done; d ready on first iter

        // 3a: Afull tile: afull[cc][u][v] = sum_r w3[c,r]*d[r,u,v] + b3[c] + A[u,v]
        for (int idx = tid; idx < 16 * V_ * V_; idx += BLK) {
            const int cc = idx / (V_ * V_);
            const int uv = idx % (V_ * V_);
            const int u = uv / V_, v = uv % V_;
            const int c = ct * 16 + cc;
            float acc = b3[c] + Ag[uv];
#pragma unroll
            for (int r = 0; r < REL; ++r)
                acc += w3[c * REL + r] * sm[OFF_D + r * (V_ * V_) + uv];
            sm[OFF_AF + cc * (UP * VP) + u * VP + v] = acc;
        }

        // 3b: GEMM1 (WMMA f32 16x16x4): inputs[cc][t,v] = wr[c,:] @ x[n,:,t,v] + br[c]
        for (int tile = wave; tile < TV / 16; tile += NWAVE) {
            const int col = tile * 16 + lo;          // tv column for this lane
            v8f acc = {};
#pragma unroll
            for (int k0 = 0; k0 < CIN; k0 += 4) {
                const int i0 = k0 + 2 * hi;
                v2f a, b;
                a.x = sm[OFF_WR + (ct * 16 + lo) * WRS + i0];       // A[m=lo][k]
                a.y = sm[OFF_WR + (ct * 16 + lo) * WRS + i0 + 1];
                b.x = xg[i0 * TV + col];                            // B[k][n=lo] (L2-resident)
                b.y = xg[(i0 + 1) * TV + col];
                acc = __builtin_amdgcn_wmma_f32_16x16x4_f32(
                        false, a, false, b, (short)0, acc, false, false);
            }
            const int t = col / V_, v = col % V_;
#pragma unroll
            for (int j = 0; j < 8; ++j) {
                const int m = j + 8 * hi;            // channel within tile
                sm[OFF_INP + m * (T_ * VP) + t * VP + v] = acc[j] + br[ct * 16 + m];
            }
        }
        __syncthreads();

        // 3c: GEMM2 (WMMA): out[t][u] = sum_v inputs[c][t][v] * Afull[c][u][v]
        {
            const int cl = wave;                     // one channel per wave
            const int c = ct * 16 + cl;
#pragma unroll
            for (int mt = 0; mt < 4; ++mt) {
#pragma unroll
                for (int nt = 0; nt < 2; ++nt) {
                    v8f acc = {};
#pragma unroll
                    for (int k0 = 0; k0 < VP; k0 += 4) {
                        const int kk = k0 + 2 * hi;
                        v2f a, b;
                        a.x = sm[OFF_INP + cl * (T_ * VP) + (mt * 16 + lo) * VP + kk];
                        a.y = sm[OFF_INP + cl * (T_ * VP) + (mt * 16 + lo) * VP + kk + 1];
                        b.x = sm[OFF_AF + cl * (UP * VP) + (nt * 16 + lo) * VP + kk];
                        b.y = sm[OFF_AF + cl * (UP * VP) + (nt * 16 + lo) * VP + kk + 1];
                        acc = __builtin_amdgcn_wmma_f32_16x16x4_f32(
                                false, a, false, b, (short)0, acc, false, false);
                    }
                    const int u = nt * 16 + lo;
                    if (u < V_) {
#pragma unroll
                        for (int j = 0; j < 8; ++j) {
                            const int t = mt * 16 + j + 8 * hi;
                            out[(((size_t)n * COUT + c) * T_ + t) * V_ + u] = acc[j];
                        }
                    }
                }
            }
        }
    }
}

extern "C" void kernel_launch(void* const* d_in, const int* in_sizes, int n_in,
                              void* d_out, int out_size, void* d_ws, size_t ws_size,
                              hipStream_t stream) {
    const float* x  = (const float*)d_in[0];
    const float* A  = (const float*)d_in[1];
    const float* w1 = (const float*)d_in[2];
    const float* b1 = (const float*)d_in[3];
    const float* w2 = (const float*)d_in[4];
    const float* b2 = (const float*)d_in[5];
    const float* w3 = (const float*)d_in[6];
    const float* b3 = (const float*)d_in[7];
    const float* wr = (const float*)d_in[8];
    const float* br = (const float*)d_in[9];
    float* out = (float*)d_out;

    const size_t shmem = (size_t)SMEM_FLOATS * sizeof(float);  // ~217 KB < 320 KB/WGP
    gcn_fused_kernel<<<dim3(N_), dim3(BLK), shmem, stream>>>(
        x, A, w1, b1, w2, b2, w3, b3, wr, br, out);
}